// GNN_24369644438205
// MI455X (gfx1250) — compile-verified
//
#include <hip/hip_runtime.h>

#define N_NODES 50000
#define N_EDGES 800000
#define D_IN    128
#define D_HID   256

typedef __attribute__((ext_vector_type(2))) float v2f;
typedef __attribute__((ext_vector_type(8))) float v8f;

// ---------------- degree / normalization ----------------

__global__ void deg_init_kernel(float* __restrict__ deg, int n) {
  int i = blockIdx.x * blockDim.x + threadIdx.x;
  if (i < n) deg[i] = 1.0f;  // self loop contributes 1 to every node's degree
}

__global__ void deg_edge_kernel(const int* __restrict__ col, float* __restrict__ deg, int e) {
  int i = blockIdx.x * blockDim.x + threadIdx.x;
  if (i < e) atomicAdd(&deg[col[i]], 1.0f);
}

__global__ void dinv_kernel(const float* __restrict__ deg, float* __restrict__ dinv, int n) {
  int i = blockIdx.x * blockDim.x + threadIdx.x;
  if (i < n) dinv[i] = rsqrtf(deg[i]);
}

// ---------------- fp32 WMMA GEMM: H[N,256] = reluOpt(A[N,din]) @ W[din,256] ----------------
// One wave computes a 16x64 output tile (4 accumulators of 16x16).
// Block = 128 threads = 4 waves covering all 256 output columns of one 16-row tile.
// A fragment (16x4 f32, 2 VGPRs): lane<16 holds K={0,1}, lane>=16 holds K={2,3}, M = lane&15.
// B fragment (4x16 f32, 2 VGPRs): lane<16 holds K={0,1} rows, lane>=16 K={2,3}, N = lane&15.
// C/D (16x16 f32, 8 VGPRs): vgpr r -> row r + 8*(lane>=16), col = lane&15.
template <int RELU>
__global__ __launch_bounds__(128) void gemm_wmma_kernel(const float* __restrict__ A,
                                                        const float* __restrict__ W,
                                                        float* __restrict__ H,
                                                        int din) {
  const int lane = threadIdx.x & 31;
  const int wave = threadIdx.x >> 5;
  const int half = lane >> 4;   // which half-wave: selects K pair
  const int mrow = lane & 15;   // M for A, N for B/C/D
  const int m_base = blockIdx.x * 16;
  const int c_base = wave * 64;

  const float* __restrict__ Arow = A + (size_t)(m_base + mrow) * din;

  v8f acc0 = {}, acc1 = {}, acc2 = {}, acc3 = {};

  for (int k = 0; k < din; k += 4) {
    const int kk = k + (half << 1);
    // kk is even and rows are 128/256-float aligned -> 8B-aligned b64 load
    v2f a = *(const v2f*)(Arow + kk);
    if (RELU) { a.x = fmaxf(a.x, 0.0f); a.y = fmaxf(a.y, 0.0f); }

    const float* __restrict__ Bp = W + (size_t)kk * 256 + c_base + mrow;
    v2f b0; b0.x = Bp[0];  b0.y = Bp[256 + 0];
    v2f b1; b1.x = Bp[16]; b1.y = Bp[256 + 16];
    v2f b2; b2.x = Bp[32]; b2.y = Bp[256 + 32];
    v2f b3; b3.x = Bp[48]; b3.y = Bp[256 + 48];

    acc0 = __builtin_amdgcn_wmma_f32_16x16x4_f32(false, a, false, b0, (short)0, acc0, false, false);
    acc1 = __builtin_amdgcn_wmma_f32_16x16x4_f32(false, a, false, b1, (short)0, acc1, false, false);
    acc2 = __builtin_amdgcn_wmma_f32_16x16x4_f32(false, a, false, b2, (short)0, acc2, false, false);
    acc3 = __builtin_amdgcn_wmma_f32_16x16x4_f32(false, a, false, b3, (short)0, acc3, false, false);
  }

  float* __restrict__ Hp = H + (size_t)(m_base + half * 8) * 256 + c_base + mrow;
#pragma unroll
  for (int r = 0; r < 8; ++r) {
    Hp[(size_t)r * 256 + 0]  = acc0[r];
    Hp[(size_t)r * 256 + 16] = acc1[r];
    Hp[(size_t)r * 256 + 32] = acc2[r];
    Hp[(size_t)r * 256 + 48] = acc3[r];
  }
}

// ---------------- aggregation ----------------

// out[i,:] = h[i,:] * dinv[i]^2   (self-loop term; also zero-initializes out)
__global__ void agg_self_kernel(const float* __restrict__ h,
                                const float* __restrict__ dinv,
                                float* __restrict__ out) {
  size_t i = (size_t)blockIdx.x * blockDim.x + threadIdx.x;  // over N*256/4 float4s
  size_t base = i * 4;
  int node = (int)(base >> 8);  // /256
  float s = dinv[node];
  s = s * s;
  float4 v = *(const float4*)(h + base);
  float4 o;
  o.x = v.x * s; o.y = v.y * s; o.z = v.z * s; o.w = v.w * s;
  *(float4*)(out + base) = o;
}

// One wave per edge: out[col,:] += h[row,:] * dinv[row]*dinv[col]
__global__ __launch_bounds__(256) void agg_edge_kernel(const float* __restrict__ h,
                                                       const float* __restrict__ dinv,
                                                       const int* __restrict__ row,
                                                       const int* __restrict__ col,
                                                       float* __restrict__ out, int e) {
  int edge = blockIdx.x * (blockDim.x >> 5) + (threadIdx.x >> 5);
  int lane = threadIdx.x & 31;
  if (edge >= e) return;
  int r = row[edge];
  int c = col[edge];
  float nrm = dinv[r] * dinv[c];
  const float* __restrict__ hp = h + (size_t)r * 256;
  float* __restrict__ op = out + (size_t)c * 256;
#pragma unroll
  for (int chunk = 0; chunk < 2; ++chunk) {
    int off = chunk * 128 + lane * 4;
    float4 v = *(const float4*)(hp + off);
    atomicAdd(op + off + 0, v.x * nrm);
    atomicAdd(op + off + 1, v.y * nrm);
    atomicAdd(op + off + 2, v.z * nrm);
    atomicAdd(op + off + 3, v.w * nrm);
  }
}

// ---------------- launch ----------------

extern "C" void kernel_launch(void* const* d_in, const int* in_sizes, int n_in,
                              void* d_out, int out_size, void* d_ws, size_t ws_size,
                              hipStream_t stream) {
  const float* x     = (const float*)d_in[0];
  const int*   ei    = (const int*)d_in[1];
  const float* W_in  = (const float*)d_in[2];
  const float* W_hid = (const float*)d_in[3];
  const float* W_out = (const float*)d_in[4];
  float* out = (float*)d_out;

  float* ws   = (float*)d_ws;
  float* deg  = ws;                                  // 65536 floats
  float* dinv = ws + 65536;                          // 65536 floats
  float* bufH = ws + 2 * 65536;                      // N*256 floats (GEMM output)
  float* bufG = bufH + (size_t)N_NODES * 256;        // N*256 floats (aggregated)

  const int* row = ei;             // edge_index[0] = source
  const int* col = ei + N_EDGES;   // edge_index[1] = target

  const int T256N = (N_NODES + 255) / 256;
  const int T256E = (N_EDGES + 255) / 256;
  const int ROW_TILES = N_NODES / 16;               // 3125, exact
  const int SELF_BLKS = (N_NODES * 256 / 4) / 256;  // exact
  const int EDGE_BLKS = (N_EDGES + 7) / 8;          // 8 waves (edges) per 256-thread block

  // degree + D^{-1/2}
  deg_init_kernel<<<T256N, 256, 0, stream>>>(deg, N_NODES);
  deg_edge_kernel<<<T256E, 256, 0, stream>>>(col, deg, N_EDGES);
  dinv_kernel<<<T256N, 256, 0, stream>>>(deg, dinv, N_NODES);

  // layer 1: h = x @ W_in ; g1 = Agg(h)         (relu deferred to next gemm's load)
  gemm_wmma_kernel<0><<<ROW_TILES, 128, 0, stream>>>(x, W_in, bufH, D_IN);
  agg_self_kernel<<<SELF_BLKS, 256, 0, stream>>>(bufH, dinv, bufG);
  agg_edge_kernel<<<EDGE_BLKS, 256, 0, stream>>>(bufH, dinv, row, col, bufG, N_EDGES);

  // layer 2: h = relu(g1) @ W_hid ; g2 = Agg(h) -> staged in d_out
  gemm_wmma_kernel<1><<<ROW_TILES, 128, 0, stream>>>(bufG, W_hid, bufH, D_HID);
  agg_self_kernel<<<SELF_BLKS, 256, 0, stream>>>(bufH, dinv, out);
  agg_edge_kernel<<<EDGE_BLKS, 256, 0, stream>>>(bufH, dinv, row, col, out, N_EDGES);

  // layer 3: h = relu(g2) @ W_out ; out = Agg(h)  (no relu on output)
  gemm_wmma_kernel<1><<<ROW_TILES, 128, 0, stream>>>(out, W_out, bufH, D_HID);
  agg_self_kernel<<<SELF_BLKS, 256, 0, stream>>>(bufH, dinv, out);
  agg_edge_kernel<<<EDGE_BLKS, 256, 0, stream>>>(bufH, dinv, row, col, out, N_EDGES);
}